// GConv_89292370084351
// MI455X (gfx1250) — compile-verified
//
#include <hip/hip_runtime.h>
#include <hip/hip_bf16.h>

#define DIN 128
#define DH  64

typedef __attribute__((ext_vector_type(2))) float v2f;
typedef __attribute__((ext_vector_type(8))) float v8f;

// ---------------- degree / normalization ----------------

__global__ void k_deg_init(float* __restrict__ deg, int n) {
    int i = blockIdx.x * blockDim.x + threadIdx.x;
    if (i < n) deg[i] = 1.0f;   // self-loop contributes 1 to every node
}

__global__ void k_deg_count(const int* __restrict__ dst, float* __restrict__ deg, int e) {
    int i = blockIdx.x * blockDim.x + threadIdx.x;
    if (i < e) atomicAdd(&deg[dst[i]], 1.0f);
}

__global__ void k_deg_finalize(float* __restrict__ deg, int n) {
    int i = blockIdx.x * blockDim.x + threadIdx.x;
    if (i < n) {
        float d = deg[i];
        deg[i] = (d > 0.0f) ? rsqrtf(d) : 0.0f;   // becomes dinv
    }
}

__global__ void k_zero(float* __restrict__ p, long long n) {
    long long i = (long long)blockIdx.x * blockDim.x + threadIdx.x;
    if (i < n) p[i] = 0.0f;
}

// ---------------- dense GEMM: h[M x 64] = x[M x K] @ W[K x 64] ----------------
// One block = 256 threads = 8 waves; each wave computes a 16-row x 64-col strip
// with 4 f32 WMMA accumulators (v_wmma_f32_16x16x4_f32).
//
// W is staged in LDS in WMMA-ready pairs: sW2[kpair][col] = {W[2k][c], W[2k+1][c]},
// so each B operand is one aligned ds_load_b64. Row stride = 80 float2: the
// row->row bank shift is 2*80 mod 64 = 32 banks, so lanes 0-15 (k-pair r) and
// lanes 16-31 (k-pair r+1) hit disjoint bank halves -> conflict-free.
//
// A-rows are clamped (not predicated) so EXEC stays all-1s for the WMMA ops;
// out-of-range rows are simply not stored.

#define BSTRIDE (DH + 16)   // row stride in float2 units (80)

__global__ void __launch_bounds__(256) k_gemm_wmma(const float* __restrict__ x,
                                                   const float* __restrict__ W,
                                                   float* __restrict__ h,
                                                   int M, int K) {
    __shared__ float sW[(DIN / 2) * BSTRIDE * 2];   // 64 * 80 * 2 floats = 40 KB

    const int tid = threadIdx.x;
    for (int i = tid; i < K * DH; i += 256) {
        int k = i / DH, c = i - k * DH;
        sW[(((k >> 1) * BSTRIDE) + c) * 2 + (k & 1)] = W[i];
    }
    __syncthreads();

    const int wave = tid >> 5;
    const int lane = tid & 31;
    const int half = lane >> 4;                // 0: lanes 0-15, 1: lanes 16-31
    const int ln   = lane & 15;

    const int rowBase = blockIdx.x * 128 + wave * 16;
    int arow = rowBase + ln;
    if (arow >= M) arow = M - 1;               // clamp: keeps EXEC uniform
    const float* xrow = x + (size_t)arow * K + 2 * half;

    v8f acc0 = {}, acc1 = {}, acc2 = {}, acc3 = {};

    for (int k0 = 0; k0 < K; k0 += 4) {
        // A 16x4 f32: lane(<16): M=ln, K={k0,k0+1}; lane(>=16): M=ln, K={k0+2,k0+3}
        v2f a;
        a.x = xrow[k0];
        a.y = xrow[k0 + 1];

        // B 4x16 f32 tiles: lanes 0-15 need k-pair (k0/2), lanes 16-31 k-pair (k0/2)+1
        const float* bp = &sW[(((k0 >> 1) + half) * BSTRIDE + ln) * 2];
        v2f b0 = *(const v2f*)(bp +  0);   // cols  0..15
        v2f b1 = *(const v2f*)(bp + 32);   // cols 16..31
        v2f b2 = *(const v2f*)(bp + 64);   // cols 32..47
        v2f b3 = *(const v2f*)(bp + 96);   // cols 48..63

        acc0 = __builtin_amdgcn_wmma_f32_16x16x4_f32(false, a, false, b0, (short)0, acc0, false, false);
        acc1 = __builtin_amdgcn_wmma_f32_16x16x4_f32(false, a, false, b1, (short)0, acc1, false, false);
        acc2 = __builtin_amdgcn_wmma_f32_16x16x4_f32(false, a, false, b2, (short)0, acc2, false, false);
        acc3 = __builtin_amdgcn_wmma_f32_16x16x4_f32(false, a, false, b3, (short)0, acc3, false, false);
    }

    // C/D layout: component i -> row (i + 8*half), column = ln within 16-col tile
    #pragma unroll
    for (int i = 0; i < 8; ++i) {
        int r = rowBase + i + 8 * half;
        if (r < M) {
            float* hr = h + (size_t)r * DH + ln;
            hr[ 0] = acc0[i];
            hr[16] = acc1[i];
            hr[32] = acc2[i];
            hr[48] = acc3[i];
        }
    }
}

// ---------------- sparse aggregation: agg[dst] += h[src] * norm ----------------
// 16 threads per edge (4 channels each via float4). Edges E..E+N-1 are the
// implicit self-loops (src = dst = e - E). h and agg are L2-resident (25.6 MB
// each vs 192 MB L2), so gathers and f32 atomic adds stay on-chip.
__global__ void k_aggregate(const float* __restrict__ h,
                            const int* __restrict__ src,
                            const int* __restrict__ dst,
                            const float* __restrict__ dinv,
                            float* __restrict__ agg,
                            int E, int n) {
    long long gid = (long long)blockIdx.x * blockDim.x + threadIdx.x;
    long long total = (long long)(E + n) * (DH / 4);
    if (gid >= total) return;

    int e = (int)(gid >> 4);
    int c = (int)(gid & 15) * 4;

    int s, d;
    if (e < E) { s = src[e]; d = dst[e]; }
    else       { s = e - E;  d = s;      }

    float nrm = dinv[s] * dinv[d];
    float4 hv = *(const float4*)(h + (size_t)s * DH + c);

    float* ap = agg + (size_t)d * DH + c;
    atomicAdd(ap + 0, hv.x * nrm);
    atomicAdd(ap + 1, hv.y * nrm);
    atomicAdd(ap + 2, hv.z * nrm);
    atomicAdd(ap + 3, hv.w * nrm);
}

// ---------------- bias + PReLU ----------------
__global__ void k_bias_prelu(const float* __restrict__ agg,
                             const float* __restrict__ b,
                             const float* __restrict__ alpha,
                             float* __restrict__ out,
                             long long total) {
    long long i = (long long)blockIdx.x * blockDim.x + threadIdx.x;
    if (i >= total) return;
    int c = (int)(i & (DH - 1));
    float z = agg[i] + b[c];
    out[i] = (z >= 0.0f) ? z : alpha[c] * z;
}

// ---------------- launch ----------------
extern "C" void kernel_launch(void* const* d_in, const int* in_sizes, int n_in,
                              void* d_out, int out_size, void* d_ws, size_t ws_size,
                              hipStream_t stream) {
    const float* x     = (const float*)d_in[0];
    const int*   ei    = (const int*)  d_in[1];
    const float* W1    = (const float*)d_in[2];
    const float* b1    = (const float*)d_in[3];
    const float* W2    = (const float*)d_in[4];
    const float* b2    = (const float*)d_in[5];
    const float* alpha = (const float*)d_in[6];

    const int N = in_sizes[0] / DIN;
    const int E = in_sizes[1] / 2;
    const int* src = ei;
    const int* dst = ei + E;

    float* out  = (float*)d_out;
    float* ws   = (float*)d_ws;
    float* dinv = ws;                       // N floats
    float* h    = dinv + N;                 // N*DH floats
    float* agg  = h + (size_t)N * DH;       // N*DH floats

    const long long ND   = (long long)N * DH;
    const long long AGGT = (long long)(E + N) * (DH / 4);

    dim3 blk(256);
    auto nb = [](long long work) { return dim3((unsigned)((work + 255) / 256)); };

    // normalization: deg -> dinv (in place)
    k_deg_init    <<<nb(N), blk, 0, stream>>>(dinv, N);
    k_deg_count   <<<nb(E), blk, 0, stream>>>(dst, dinv, E);
    k_deg_finalize<<<nb(N), blk, 0, stream>>>(dinv, N);

    const unsigned gemmBlocks = (unsigned)((N + 127) / 128);

    // ---- layer 1 ----
    k_gemm_wmma <<<dim3(gemmBlocks), blk, 0, stream>>>(x, W1, h, N, DIN);
    k_zero      <<<nb(ND),   blk, 0, stream>>>(agg, ND);
    k_aggregate <<<nb(AGGT), blk, 0, stream>>>(h, src, dst, dinv, agg, E, N);
    k_bias_prelu<<<nb(ND),   blk, 0, stream>>>(agg, b1, alpha, h, ND);   // z1 -> h

    // ---- layer 2 ----
    k_gemm_wmma <<<dim3(gemmBlocks), blk, 0, stream>>>(h, W2, agg, N, DH); // h2 -> agg
    k_zero      <<<nb(ND),   blk, 0, stream>>>(out, ND);
    k_aggregate <<<nb(AGGT), blk, 0, stream>>>(agg, src, dst, dinv, out, E, N);
    k_bias_prelu<<<nb(ND),   blk, 0, stream>>>(out, b2, alpha, out, ND);  // in place
}